// GraphConv_56908316672595
// MI455X (gfx1250) — compile-verified
//
#include <hip/hip_runtime.h>

typedef __attribute__((ext_vector_type(2))) float v2f;
typedef __attribute__((ext_vector_type(8))) float v8f;

#define N_NODES 20000
#define N_EDGES 640000
#define D 256

// ---------------------------------------------------------------------------
// Kernel 1: zero the aggregation buffer (d_ws is poisoned by the harness).
// 5,120,000 floats -> 1,280,000 float4 stores -> 5000 blocks x 256 threads.
// ---------------------------------------------------------------------------
__global__ void __launch_bounds__(256) graphconv_zero_agg(float4* __restrict__ agg4) {
    const size_t i = (size_t)blockIdx.x * 256 + threadIdx.x;
    agg4[i] = make_float4(0.f, 0.f, 0.f, 0.f);
}

// ---------------------------------------------------------------------------
// Kernel 2: edge scatter. msgs = x[src] * w ; agg[dst] += msgs.
// 64 threads per edge, float4 per thread (global_load_b128 gather,
// 4x global_atomic_add_f32 scatter). x and agg are L2-resident (20.5 MB each).
// ---------------------------------------------------------------------------
__global__ void __launch_bounds__(256) graphconv_scatter(
    const float* __restrict__ x,
    const int*   __restrict__ edge_index,   // [2, N_EDGES]
    const float* __restrict__ edge_attr,    // [N_EDGES]
    float*       __restrict__ agg)          // [N_NODES, D]
{
    const int tid  = threadIdx.x;
    const int edge = blockIdx.x * 4 + (tid >> 6);   // 4 edges per 256-thread block
    const int d4   = tid & 63;                      // float4 slot within D=256

    const int   src = edge_index[edge];
    const int   dst = edge_index[N_EDGES + edge];
    const float w   = edge_attr[edge];

    const float4 v = ((const float4*)(x + (size_t)src * D))[d4];
    float* a = agg + (size_t)dst * D + (size_t)d4 * 4;
    atomicAdd(a + 0, v.x * w);
    atomicAdd(a + 1, v.y * w);
    atomicAdd(a + 2, v.z * w);
    atomicAdd(a + 3, v.w * w);
}

// ---------------------------------------------------------------------------
// Kernel 3: out = agg @ W_rel^T + b_rel + x @ W_root^T, via V_WMMA_F32_16X16X4_F32.
// One wave32 per 16x16 output tile. 8 waves/block, 2500 blocks = 20000 tiles
// (= 1250 M-tiles x 16 N-tiles, exact: 20000 % 16 == 0). EXEC is all-ones
// (no divergence), as WMMA requires.
//
// VGPR layouts (ISA 7.12.2, 32-bit, wave32):
//   A 16x4 : lane%16 = M row, lane/16 selects K in {0,1} vs {2,3} (2 VGPRs)
//   B 4x16 : lane%16 = N col, lane/16 selects K in {0,1} vs {2,3} (2 VGPRs)
//   C 16x16: VGPR i -> row (i + 8*(lane/16)), col lane%16
// y = a @ W^T  =>  B[k][n] = W[n*D + k]  (coalesced b64 row reads of W).
// ---------------------------------------------------------------------------
__global__ void __launch_bounds__(256) graphconv_gemm(
    const float* __restrict__ agg,
    const float* __restrict__ x,
    const float* __restrict__ W_rel,
    const float* __restrict__ b_rel,
    const float* __restrict__ W_root,
    float*       __restrict__ out)
{
    const int lane = threadIdx.x & 31;
    const int wave = threadIdx.x >> 5;
    const int tile = blockIdx.x * 8 + wave;     // 0..19999
    const int mt   = tile >> 4;                 // 0..1249 (M tile)
    const int nt   = tile & 15;                 // 0..15   (N tile)
    const int half = lane >> 4;                 // 0/1 -> K sub-pair
    const int l16  = lane & 15;

    const int m  = mt * 16 + l16;               // this lane's A row
    const int n  = nt * 16 + l16;               // this lane's B/C column
    const int kb = half * 2;                    // K offset within 4-wide step

    v8f acc = {};

    // Pass 1: agg @ W_rel^T
    {
        const float* Arow = agg   + (size_t)m * D + kb;
        const float* Brow = W_rel + (size_t)n * D + kb;
        #pragma unroll 8
        for (int k0 = 0; k0 < D; k0 += 4) {
            v2f a = *(const v2f*)(Arow + k0);
            v2f b = *(const v2f*)(Brow + k0);
            acc = __builtin_amdgcn_wmma_f32_16x16x4_f32(
                /*neg_a=*/false, a, /*neg_b=*/false, b,
                /*c_mod=*/(short)0, acc, /*reuse_a=*/false, /*reuse_b=*/false);
        }
    }
    // Pass 2: + x @ W_root^T (accumulates into same C)
    {
        const float* Arow = x      + (size_t)m * D + kb;
        const float* Brow = W_root + (size_t)n * D + kb;
        #pragma unroll 8
        for (int k0 = 0; k0 < D; k0 += 4) {
            v2f a = *(const v2f*)(Arow + k0);
            v2f b = *(const v2f*)(Brow + k0);
            acc = __builtin_amdgcn_wmma_f32_16x16x4_f32(
                false, a, false, b, (short)0, acc, false, false);
        }
    }

    // Bias + store (C layout: VGPR i -> row base + i, col n)
    const float bias  = b_rel[n];
    const int   rbase = mt * 16 + half * 8;
    #pragma unroll
    for (int i = 0; i < 8; ++i) {
        out[(size_t)(rbase + i) * D + n] = acc[i] + bias;
    }
}

// ---------------------------------------------------------------------------
extern "C" void kernel_launch(void* const* d_in, const int* in_sizes, int n_in,
                              void* d_out, int out_size, void* d_ws, size_t ws_size,
                              hipStream_t stream) {
    const float* x      = (const float*)d_in[0];
    const int*   eidx   = (const int*)  d_in[1];
    const float* eattr  = (const float*)d_in[2];
    const float* W_rel  = (const float*)d_in[3];
    const float* b_rel  = (const float*)d_in[4];
    const float* W_root = (const float*)d_in[5];
    float*       out    = (float*)d_out;
    float*       agg    = (float*)d_ws;   // N_NODES * D floats = 20.48 MB

    graphconv_zero_agg<<<(N_NODES * D / 4) / 256, 256, 0, stream>>>((float4*)agg);
    graphconv_scatter <<<N_EDGES / 4,              256, 0, stream>>>(x, eidx, eattr, agg);
    graphconv_gemm    <<<(N_NODES / 16 * 16) / 8,  256, 0, stream>>>(agg, x, W_rel, b_rel, W_root, out);
}